// Model_15307263443703
// MI455X (gfx1250) — compile-verified
//
#include <hip/hip_runtime.h>
#include <hip/hip_bf16.h>

// ---------------------------------------------------------------------------
// Scene-change detector for 8K frames (4320x7680 f32).
// Outputs: [is_scene_change (0/1), sad_score, chi_sq]  (3 floats)
//
// Design (MI455X / gfx1250, wave32):
//  * single streaming pass, float4 non-temporal loads (265 MB, HBM-bound)
//  * per-wave LDS histograms (integer ds atomics, deterministic)
//  * per-block float partials -> fixed-order reduction (deterministic replay)
//  * finalize: cross-lane sums via V_WMMA_F32_16X16X4_F32 with B = ones
//    (D[m][n] = sum_k A[m][k]; summing D rows + lane0/lane16 gives the sum
//     of all 64 A register slots regardless of A's (m,k) slot mapping)
// ---------------------------------------------------------------------------

typedef __attribute__((ext_vector_type(2))) float v2f;
typedef __attribute__((ext_vector_type(4))) float v4f;
typedef __attribute__((ext_vector_type(8))) float v8f;

#define BINS 32
#define TPB  256   // 8 waves of 32
#define WAVES_PER_BLOCK (TPB / 32)

// ---------------------------------------------------------------------------
__global__ void sc_init_bins(unsigned int* gbins) {
    int t = threadIdx.x;
    if (t < 2 * BINS) gbins[t] = 0u;
}

// ---------------------------------------------------------------------------
__global__ void sc_main(const float* __restrict__ f1,
                        const float* __restrict__ f2,
                        float* __restrict__ blk_sad,
                        float* __restrict__ blk_abs,
                        unsigned int* __restrict__ gbins,
                        int n) {
    __shared__ unsigned int whist[WAVES_PER_BLOCK][2 * BINS];
    __shared__ float redS[TPB];
    __shared__ float redA[TPB];

    const int tid  = threadIdx.x;
    const int wid  = tid >> 5;
    const int gtid = blockIdx.x * TPB + tid;
    const int nthreads = gridDim.x * TPB;

    // zero per-wave histograms (512 u32 over 256 threads)
    whist[tid >> 6][tid & 63] = 0u;
    whist[(tid >> 6) + 4][tid & 63] = 0u;
    __syncthreads();

    float sadAcc = 0.0f;
    float absAcc = 0.0f;

    const int n4 = n >> 2;
    const v4f* __restrict__ p1 = (const v4f*)f1;
    const v4f* __restrict__ p2 = (const v4f*)f2;

    for (int i = gtid; i < n4; i += nthreads) {
        v4f a = __builtin_nontemporal_load(&p1[i]);   // global_load_b128 th:NT
        v4f b = __builtin_nontemporal_load(&p2[i]);
#pragma unroll
        for (int c = 0; c < 4; ++c) {
            float x = a[c];
            float y = b[c];
            sadAcc += __builtin_fabsf(x - y);
            absAcc += __builtin_fabsf(x);
            int q1 = (int)(x * 31.0f);
            q1 = q1 < 0 ? 0 : (q1 > 31 ? 31 : q1);
            int q2 = (int)(y * 31.0f);
            q2 = q2 < 0 ? 0 : (q2 > 31 ? 31 : q2);
            atomicAdd(&whist[wid][q1], 1u);           // ds_add_u32 (no rtn)
            atomicAdd(&whist[wid][BINS + q2], 1u);
        }
    }
    // scalar tail (n not divisible by 4)
    for (int i = (n4 << 2) + gtid; i < n; i += nthreads) {
        float x = f1[i];
        float y = f2[i];
        sadAcc += __builtin_fabsf(x - y);
        absAcc += __builtin_fabsf(x);
        int q1 = (int)(x * 31.0f);
        q1 = q1 < 0 ? 0 : (q1 > 31 ? 31 : q1);
        int q2 = (int)(y * 31.0f);
        q2 = q2 < 0 ? 0 : (q2 > 31 ? 31 : q2);
        atomicAdd(&whist[wid][q1], 1u);
        atomicAdd(&whist[wid][BINS + q2], 1u);
    }

    // deterministic fixed-order block reduction of the float partials
    redS[tid] = sadAcc;
    redA[tid] = absAcc;
    __syncthreads();
#pragma unroll
    for (int s = TPB / 2; s > 0; s >>= 1) {
        if (tid < s) {
            redS[tid] += redS[tid + s];
            redA[tid] += redA[tid + s];
        }
        __syncthreads();
    }
    if (tid == 0) {
        blk_sad[blockIdx.x] = redS[0];
        blk_abs[blockIdx.x] = redA[0];
    }

    // fold the 8 per-wave histograms, push to global (integer atomics: exact,
    // order-independent -> deterministic across graph replays)
    if (tid < 2 * BINS) {
        unsigned int v = 0;
#pragma unroll
        for (int w = 0; w < WAVES_PER_BLOCK; ++w) v += whist[w][tid];
        atomicAdd(&gbins[tid], v);
    }
}

// ---------------------------------------------------------------------------
// Sum of all 64 register slots (2 VGPRs x 32 lanes) via one WMMA with B=ones.
// Requires EXEC == all ones (single full wave, no divergence at call site).
__device__ __forceinline__ float wave_sum64_wmma(float a0, float a1) {
    v2f A; A[0] = a0; A[1] = a1;
    v2f B; B[0] = 1.0f; B[1] = 1.0f;
    v8f C = {};
    v8f D = __builtin_amdgcn_wmma_f32_16x16x4_f32(
        /*neg_a=*/false, A, /*neg_b=*/false, B,
        /*c_mod=*/(short)0, C, /*reuse_a=*/false, /*reuse_b=*/false);
    // lane 0 holds D[0..7][0] = rowsums 0..7 ; lane 16 holds D[8..15][0]
    float s = D[0] + D[1] + D[2] + D[3] + D[4] + D[5] + D[6] + D[7];
    return __shfl(s, 0, 32) + __shfl(s, 16, 32);
}

__global__ void sc_finalize(const float* __restrict__ blk_sad,
                            const float* __restrict__ blk_abs,
                            const unsigned int* __restrict__ gbins,
                            float* __restrict__ out,
                            int nb, int n) {
    const int lane = threadIdx.x;   // 32 threads = one wave

    float s0 = 0.0f, s1 = 0.0f, a0 = 0.0f, a1 = 0.0f;
    for (int base = 0; base < nb; base += 64) {
        int i0 = base + lane;
        int i1 = base + 32 + lane;
        float vs0 = (i0 < nb) ? blk_sad[i0] : 0.0f;
        float va0 = (i0 < nb) ? blk_abs[i0] : 0.0f;
        float vs1 = (i1 < nb) ? blk_sad[i1] : 0.0f;
        float va1 = (i1 < nb) ? blk_abs[i1] : 0.0f;
        s0 += vs0; a0 += va0;
        s1 += vs1; a1 += va1;
    }

    float sadTot = wave_sum64_wmma(s0, s1);
    float absTot = wave_sum64_wmma(a0, a1);

    const float inv = 1.0f / (float)n;   // n < 2^25 -> exact in f32

    // chi-square of normalized 32-bin histograms: one term per lane
    unsigned int c1 = gbins[lane];
    unsigned int c2 = gbins[BINS + lane];
    float h1 = (float)c1 * inv;
    float h2 = (float)c2 * inv;
    float d  = h1 - h2;
    float term = (d * d) / (h1 + h2 + 1e-10f);
    float chi = wave_sum64_wmma(term, 0.0f) * 0.5f;

    if (lane == 0) {
        float sadMean = sadTot * inv;
        float absMean = absTot * inv;
        float sadScore = sadMean / fmaxf(absMean, 1e-6f);
        out[0] = (sadScore > 0.3f || chi > 0.5f) ? 1.0f : 0.0f;
        out[1] = sadScore;
        out[2] = chi;
    }
}

// ---------------------------------------------------------------------------
extern "C" void kernel_launch(void* const* d_in, const int* in_sizes, int n_in,
                              void* d_out, int out_size, void* d_ws, size_t ws_size,
                              hipStream_t stream) {
    (void)n_in; (void)out_size;
    const float* f1 = (const float*)d_in[0];
    const float* f2 = (const float*)d_in[1];
    float* out = (float*)d_out;
    const int n = in_sizes[0];

    // blocks: 4050 * 256 threads * 8 float4 = exactly 33,177,600 elements,
    // clamped so the partial arrays always fit in the provided workspace.
    int nb = 4050;
    if (ws_size < (size_t)(2 * nb * sizeof(float) + 2 * BINS * sizeof(unsigned int))) {
        size_t avail = ws_size > 2 * BINS * sizeof(unsigned int)
                           ? ws_size - 2 * BINS * sizeof(unsigned int) : 0;
        int cap = (int)(avail / (2 * sizeof(float)));
        nb = cap < 1 ? 1 : (cap < nb ? cap : nb);
    }

    float* blk_sad = (float*)d_ws;
    float* blk_abs = blk_sad + nb;
    unsigned int* gbins = (unsigned int*)(blk_abs + nb);

    sc_init_bins<<<1, 64, 0, stream>>>(gbins);
    sc_main<<<nb, TPB, 0, stream>>>(f1, f2, blk_sad, blk_abs, gbins, n);
    sc_finalize<<<1, 32, 0, stream>>>(blk_sad, blk_abs, gbins, out, nb, n);
}